// BeliefPropagation_111669149760
// MI455X (gfx1250) — compile-verified
//
#include <hip/hip_runtime.h>
#include <hip/hip_bf16.h>
#include <stdint.h>

#define NN 50000
#define EE 800000
#define ITERS 10
#define NPB 32          // nodes per block in node-product kernel (256 thr / 8 q)
#define STAGE_CAP 4096  // LDS staging capacity for edge ids (16KB)

#define NB_E 3125       // ceil(EE/256)
#define NB_N8 1563      // ceil(NN*8/256)
#define NB_NODE 1563    // ceil(NN/NPB)
#define NB_N 196        // ceil(NN/256)

typedef float v2f __attribute__((ext_vector_type(2)));
typedef float v8f __attribute__((ext_vector_type(8)));

// ---------------------------------------------------------------------------
// Block-level column sum over 8 columns (column = tid&7). Each of the 8 waves
// reduces its 32 lanes to per-q sums; a WMMA f32 16x16x4 with A=ones performs
// the lane(l) + lane(l+16) pair-sum exactly in f32 (B[0,n]=c_n, B[2,n]=c_{n+16},
// so D[*,n] = c_n + c_{n+16}); one xor-8 shuffle folds the q/q+8 groups.
// Deterministic (fixed order everywhere). Requires full EXEC (no early returns).
// ---------------------------------------------------------------------------
__device__ __forceinline__ void block_col_sum(float c, float* s_wav, float* out8) {
  int tid = threadIdx.x;
  float t;
#if __has_builtin(__builtin_amdgcn_wmma_f32_16x16x4_f32)
  {
    v2f a; a[0] = 1.0f; a[1] = 1.0f;
    v2f b; b[0] = c;    b[1] = 0.0f;
    v8f acc = {};
    acc = __builtin_amdgcn_wmma_f32_16x16x4_f32(false, a, false, b, (short)0,
                                                acc, false, false);
    t = acc[0];                    // = c[l&15] + c[(l&15)+16]
    t += __shfl_xor(t, 8, 32);     // fold q and q+8 groups
  }
#else
  t = c;
  t += __shfl_xor(t, 16, 32);
  t += __shfl_xor(t, 8, 32);
#endif
  int lane = tid & 31, wid = tid >> 5;
  if (lane < 8) s_wav[wid * 8 + lane] = t;
  __syncthreads();
  if (tid < 8) {
    float s = 0.0f;
    for (int w = 0; w < 8; ++w) s += s_wav[w * 8 + tid];  // fixed order
    out8[tid] = s;
  }
  __syncthreads();
}

// ---- setup kernels ---------------------------------------------------------

__global__ void __launch_bounds__(256) k_ew(const float* __restrict__ beta,
                                            const float* __restrict__ w,
                                            float* __restrict__ ew,
                                            float* __restrict__ part, int n) {
  __shared__ float s_w[8];
  int i = blockIdx.x * 256 + threadIdx.x;
  float v = 0.0f;
  if (i < n) {
    float wi = w[i];
    v = wi;
    ew[i] = expf((*beta) * wi) - 1.0f;
  }
  for (int m = 16; m >= 1; m >>= 1) v += __shfl_xor(v, m, 32);
  int lane = threadIdx.x & 31, wid = threadIdx.x >> 5;
  if (lane == 0) s_w[wid] = v;
  __syncthreads();
  if (threadIdx.x == 0) {
    float s = 0.0f;
    for (int k = 0; k < 8; ++k) s += s_w[k];  // fixed order
    part[blockIdx.x] = s;
  }
}

__global__ void k_reduce1(const float* __restrict__ part, int n,
                          float* __restrict__ out) {
  __shared__ float acc[256];
  float s = 0.0f;
  for (int i = threadIdx.x; i < n; i += 256) s += part[i];
  acc[threadIdx.x] = s;
  __syncthreads();
  if (threadIdx.x == 0) {
    float t = 0.0f;
    for (int k = 0; k < 256; ++k) t += acc[k];  // fixed order
    *out = t;
  }
}

__global__ void __launch_bounds__(256) k_colsum(const float* __restrict__ x,
                                                float* __restrict__ part,
                                                int nelem) {
  __shared__ float s_wav[64];
  int i = blockIdx.x * 256 + threadIdx.x;
  float c = (i < nelem) ? x[i] : 0.0f;
  block_col_sum(c, s_wav, part + (size_t)blockIdx.x * 8);
}

// h_q = -beta * mean_w * sum_n psi[n][q];  exph = exp(h). Fixed-order reduce.
__global__ void k_h(const float* __restrict__ beta, const float* __restrict__ wsum,
                    const float* __restrict__ part, int nparts,
                    float* __restrict__ exph) {
  __shared__ float acc[64];
  int t = threadIdx.x;           // 64 threads: q = t&7, stripe = t>>3
  int q = t & 7, st = t >> 3;
  float s = 0.0f;
  for (int p = st; p < nparts; p += 8) s += part[(size_t)p * 8 + q];
  acc[t] = s;
  __syncthreads();
  if (t < 8) {
    float tot = 0.0f;
    for (int k = 0; k < 8; ++k) tot += acc[k * 8 + t];  // fixed order
    float mw = (*wsum) / ((float)NN * (float)NN);
    exph[t] = expf(-(*beta) * mw * tot);
  }
}

__global__ void __launch_bounds__(256) k_deg(const int* __restrict__ dst,
                                             int* __restrict__ deg, int n) {
  int e = blockIdx.x * 256 + threadIdx.x;
  if (e < n) atomicAdd(&deg[dst[e]], 1);
}

// single-block exclusive scan of deg[0..n) -> row_ptr[0..n]
__global__ void __launch_bounds__(1024) k_scan(const int* __restrict__ deg,
                                               int* __restrict__ row_ptr, int n) {
  __shared__ int wsums[32];
  __shared__ int s_run;
  int tid = threadIdx.x;
  int lane = tid & 31, wid = tid >> 5;
  if (tid == 0) s_run = 0;
  __syncthreads();
  for (int base = 0; base < n; base += 1024) {
    int i = base + tid;
    int v = (i < n) ? deg[i] : 0;
    int incl = v;
    for (int d = 1; d < 32; d <<= 1) {
      int x = __shfl_up(incl, d, 32);
      if (lane >= d) incl += x;
    }
    if (lane == 31) wsums[wid] = incl;
    __syncthreads();
    if (wid == 0) {
      int s = wsums[lane];
      for (int d = 1; d < 32; d <<= 1) {
        int x = __shfl_up(s, d, 32);
        if (lane >= d) s += x;
      }
      wsums[lane] = s;
    }
    __syncthreads();
    int woff = (wid > 0) ? wsums[wid - 1] : 0;
    if (i < n) row_ptr[i] = s_run + woff + incl - v;
    __syncthreads();
    if (tid == 0) s_run += wsums[31];
    __syncthreads();
  }
  if (threadIdx.x == 0) row_ptr[n] = s_run;
}

__global__ void __launch_bounds__(256) k_copy_cursor(const int* __restrict__ rp,
                                                     int* __restrict__ cur, int n) {
  int i = blockIdx.x * 256 + threadIdx.x;
  if (i < n) cur[i] = rp[i];
}

__global__ void __launch_bounds__(256) k_scatter(const int* __restrict__ dst,
                                                 int* __restrict__ cur,
                                                 int* __restrict__ eids, int n) {
  int e = blockIdx.x * 256 + threadIdx.x;
  if (e < n) {
    int p = atomicAdd(&cur[dst[e]], 1);
    eids[p] = e;
  }
}

// per-node insertion sort of its edge-id segment -> deterministic product order
__global__ void __launch_bounds__(256) k_sortseg(int* __restrict__ eids,
                                                 const int* __restrict__ rp, int n) {
  int nd = blockIdx.x * 256 + threadIdx.x;
  if (nd >= n) return;
  int b = rp[nd], e = rp[nd + 1];
  for (int i = b + 1; i < e; ++i) {
    int key = eids[i];
    int j = i - 1;
    while (j >= b && eids[j] > key) { eids[j + 1] = eids[j]; --j; }
    eids[j + 1] = key;
  }
}

// fac = 1 + ew * msg  (rows of 8 floats, float4 pairs)
__global__ void __launch_bounds__(256) k_fac(const float* __restrict__ msg,
                                             const float* __restrict__ ew,
                                             float* __restrict__ fac, int n) {
  int e = blockIdx.x * 256 + threadIdx.x;
  if (e >= n) return;
  float w = ew[e];
  const float4* mr = (const float4*)(msg + (size_t)e * 8);
  float4 m0 = mr[0], m1 = mr[1];
  float4* fr = (float4*)(fac + (size_t)e * 8);
  fr[0] = make_float4(1.f + w * m0.x, 1.f + w * m0.y, 1.f + w * m0.z, 1.f + w * m0.w);
  fr[1] = make_float4(1.f + w * m1.x, 1.f + w * m1.y, 1.f + w * m1.z, 1.f + w * m1.w);
}

// ---------------------------------------------------------------------------
// Node product kernel: 256 threads = 32 nodes x 8 q-lanes. Each block stages
// its contiguous CSR edge-id segment into LDS with gfx1250 async loads
// (global_load_async_to_lds_b32 + s_wait_asynccnt), then 8 lanes per node
// broadcast-read edge ids from LDS and multiply coalesced 32B fac rows.
// mode=1 additionally computes psi (normalized exph*prod), writes it, and
// emits deterministic per-block psi column partial sums (WMMA-assisted).
// ---------------------------------------------------------------------------
__global__ void __launch_bounds__(256) k_nodeprod(
    const float* __restrict__ fac, const int* __restrict__ row_ptr,
    const int* __restrict__ eids, const float* __restrict__ exph,
    float* __restrict__ node_prod, float* __restrict__ psi_out,
    float* __restrict__ psi_part, int mode, int n) {
  __shared__ int s_eids[STAGE_CAP];
  __shared__ float s_wav[64];
  int tid = threadIdx.x;
  int node0 = blockIdx.x * NPB;
  int nend = node0 + NPB;
  if (nend > n) nend = n;
  int seg_start = row_ptr[node0];
  int seg_end = row_ptr[nend];
  int len = seg_end - seg_start;
  bool staged = (len <= STAGE_CAP);  // block-uniform
  if (staged) {
    unsigned long long base = (unsigned long long)eids;
    for (int i = tid; i < len; i += 256) {
#if defined(__HIP_DEVICE_COMPILE__)
      unsigned lds_off = (unsigned)(uintptr_t)(&s_eids[i]);     // low 32b = LDS offset
      unsigned goff = (unsigned)((seg_start + i) * 4);
      asm volatile("global_load_async_to_lds_b32 %0, %1, %2 offset:0"
                   :
                   : "v"(lds_off), "v"(goff), "s"(base)
                   : "memory");
#else
      s_eids[i] = eids[seg_start + i];
#endif
    }
#if defined(__HIP_DEVICE_COMPILE__)
    asm volatile("s_wait_asynccnt 0x0" ::: "memory");
#endif
  }
  __syncthreads();

  int node = node0 + (tid >> 3);
  int q = tid & 7;
  bool valid = node < n;
  int b = 0, d = 0;
  if (valid) { b = row_ptr[node]; d = row_ptr[node + 1] - b; }
  float prod = 1.0f;
  if (staged) {
    int off = b - seg_start;
    for (int j = 0; j < d; ++j) {
      int eid = s_eids[off + j];                 // LDS broadcast across 8 lanes
      prod *= fac[(size_t)eid * 8 + q];          // coalesced 32B row
    }
  } else {
    for (int j = 0; j < d; ++j) {
      int eid = eids[b + j];
      prod *= fac[(size_t)eid * 8 + q];
    }
  }
  if (valid) node_prod[(size_t)node * 8 + q] = prod;

  if (mode) {
    float v = valid ? exph[q] * prod : 0.0f;
    float rs = v;
    rs += __shfl_xor(rs, 1, 8);
    rs += __shfl_xor(rs, 2, 8);
    rs += __shfl_xor(rs, 4, 8);
    float psi = valid ? v / rs : 0.0f;           // rs>0 when valid (fac>=1)
    if (valid) psi_out[(size_t)node * 8 + q] = psi;
    block_col_sum(psi, s_wav, psi_part + (size_t)blockIdx.x * 8);
  }
}

// ---------------------------------------------------------------------------
// Message update: raw = exph * node_prod[src] / fac_old[rev]; normalize;
// write msg and next iteration's fac (fac_new = 1 + ew*new_msg) in one pass.
// Optionally track max|new-old| (last iteration only).
// ---------------------------------------------------------------------------
__global__ void __launch_bounds__(256) k_msg(
    float* __restrict__ msg, const float* __restrict__ ew,
    const int* __restrict__ src, const int* __restrict__ rev,
    const float* __restrict__ np, const float* __restrict__ fac_old,
    float* __restrict__ fac_new, const float* __restrict__ exph,
    unsigned* __restrict__ diff_out, int compute_diff, int n) {
  int e = blockIdx.x * 256 + threadIdx.x;
  float dmax = 0.0f;
  if (e < n) {
    int s = src[e];
    int r = rev[e];
    float w = ew[e];
    const float4* fr = (const float4*)(fac_old + (size_t)r * 8);
    __builtin_prefetch((const void*)fr, 0, 1);   // global_prefetch of gathered row
    float4 h0 = ((const float4*)exph)[0];
    float4 h1 = ((const float4*)exph)[1];
    const float4* nr = (const float4*)(np + (size_t)s * 8);
    float4 n0 = nr[0], n1 = nr[1];
    float4 f0 = fr[0], f1 = fr[1];
    float r0 = h0.x * n0.x / f0.x;
    float r1 = h0.y * n0.y / f0.y;
    float r2 = h0.z * n0.z / f0.z;
    float r3 = h0.w * n0.w / f0.w;
    float r4 = h1.x * n1.x / f1.x;
    float r5 = h1.y * n1.y / f1.y;
    float r6 = h1.z * n1.z / f1.z;
    float r7 = h1.w * n1.w / f1.w;
    float inv = 1.0f / (r0 + r1 + r2 + r3 + r4 + r5 + r6 + r7);
    float4* mrow = (float4*)(msg + (size_t)e * 8);
    float4 o0 = make_float4(0.f, 0.f, 0.f, 0.f), o1 = o0;
    if (compute_diff) { o0 = mrow[0]; o1 = mrow[1]; }
    float4 m0 = make_float4(r0 * inv, r1 * inv, r2 * inv, r3 * inv);
    float4 m1 = make_float4(r4 * inv, r5 * inv, r6 * inv, r7 * inv);
    mrow[0] = m0;
    mrow[1] = m1;
    float4* fn = (float4*)(fac_new + (size_t)e * 8);
    fn[0] = make_float4(1.f + w * m0.x, 1.f + w * m0.y, 1.f + w * m0.z, 1.f + w * m0.w);
    fn[1] = make_float4(1.f + w * m1.x, 1.f + w * m1.y, 1.f + w * m1.z, 1.f + w * m1.w);
    if (compute_diff) {
      dmax = fmaxf(dmax, fabsf(m0.x - o0.x));
      dmax = fmaxf(dmax, fabsf(m0.y - o0.y));
      dmax = fmaxf(dmax, fabsf(m0.z - o0.z));
      dmax = fmaxf(dmax, fabsf(m0.w - o0.w));
      dmax = fmaxf(dmax, fabsf(m1.x - o1.x));
      dmax = fmaxf(dmax, fabsf(m1.y - o1.y));
      dmax = fmaxf(dmax, fabsf(m1.z - o1.z));
      dmax = fmaxf(dmax, fabsf(m1.w - o1.w));
    }
  }
  if (compute_diff) {
    for (int m = 16; m >= 1; m >>= 1) dmax = fmaxf(dmax, __shfl_xor(dmax, m, 32));
    if ((threadIdx.x & 31) == 0) atomicMax(diff_out, __float_as_uint(dmax));
  }
}

// ---------------------------------------------------------------------------

extern "C" void kernel_launch(void* const* d_in, const int* in_sizes, int n_in,
                              void* d_out, int out_size, void* d_ws, size_t ws_size,
                              hipStream_t stream) {
  const float* beta = (const float*)d_in[0];
  const float* wgt  = (const float*)d_in[1];
  const float* msg0 = (const float*)d_in[2];
  const float* psi0 = (const float*)d_in[3];
  const int*   ei   = (const int*)d_in[4];
  const int*   rev  = (const int*)d_in[5];
  const int* src = ei;
  const int* dst = ei + EE;
  (void)in_sizes; (void)n_in; (void)out_size; (void)ws_size;

  float* out = (float*)d_out;
  float* out_msg = out;                       // [EE,8]
  float* out_psi = out + (size_t)EE * 8;      // [NN,8]
  unsigned* out_diff = (unsigned*)(out + (size_t)EE * 8 + (size_t)NN * 8);

  char* ws = (char*)d_ws;
  size_t off = 0;
  auto alloc = [&](size_t bytes) -> void* {
    void* p = ws + off;
    off = (off + bytes + 255) & ~(size_t)255;
    return p;
  };
  float* facA   = (float*)alloc((size_t)EE * 8 * 4);
  float* facB   = (float*)alloc((size_t)EE * 8 * 4);
  float* ewb    = (float*)alloc((size_t)EE * 4);
  float* np     = (float*)alloc((size_t)NN * 8 * 4);
  int*   rp     = (int*)alloc((size_t)(NN + 1) * 4);
  int*   deg    = (int*)alloc((size_t)NN * 4);      // reused as scatter cursor
  int*   eids   = (int*)alloc((size_t)EE * 4);
  float* ppart  = (float*)alloc((size_t)NB_N8 * 8 * 4);
  float* wpart  = (float*)alloc((size_t)NB_E * 4);
  float* wsum   = (float*)alloc(256);
  float* exph   = (float*)alloc(256);

  // ---- setup (once per launch; deterministic) ----
  hipMemsetAsync(deg, 0, (size_t)NN * 4, stream);
  hipMemsetAsync(out_diff, 0, 4, stream);

  k_ew<<<NB_E, 256, 0, stream>>>(beta, wgt, ewb, wpart, EE);
  k_reduce1<<<1, 256, 0, stream>>>(wpart, NB_E, wsum);
  k_colsum<<<NB_N8, 256, 0, stream>>>(psi0, ppart, NN * 8);
  k_h<<<1, 64, 0, stream>>>(beta, wsum, ppart, NB_N8, exph);   // h0 from psi_init

  k_deg<<<NB_E, 256, 0, stream>>>(dst, deg, EE);
  k_scan<<<1, 1024, 0, stream>>>(deg, rp, NN);
  k_copy_cursor<<<NB_N, 256, 0, stream>>>(rp, deg, NN);        // cursor = row_ptr
  k_scatter<<<NB_E, 256, 0, stream>>>(dst, deg, eids, EE);
  k_sortseg<<<NB_N, 256, 0, stream>>>(eids, rp, NN);

  hipMemcpyAsync(out_msg, msg0, (size_t)EE * 8 * 4, hipMemcpyDeviceToDevice, stream);
  k_fac<<<NB_E, 256, 0, stream>>>(out_msg, ewb, facA, EE);
  k_nodeprod<<<NB_NODE, 256, 0, stream>>>(facA, rp, eids, exph, np,
                                          nullptr, nullptr, 0, NN);

  // ---- 10 BP iterations ----
  for (int it = 0; it < ITERS; ++it) {
    float* fo = (it & 1) ? facB : facA;
    float* fn = (it & 1) ? facA : facB;
    k_msg<<<NB_E, 256, 0, stream>>>(out_msg, ewb, src, rev, np, fo, fn, exph,
                                    out_diff, (it == ITERS - 1) ? 1 : 0, EE);
    // product of fn doubles as node_prod for the NEXT iteration's k_msg
    k_nodeprod<<<NB_NODE, 256, 0, stream>>>(fn, rp, eids, exph, np,
                                            out_psi, ppart, 1, NN);
    k_h<<<1, 64, 0, stream>>>(beta, wsum, ppart, NB_N8, exph);
  }
}